// GcnLayers_2559800508646
// MI455X (gfx1250) — compile-verified
//
#include <hip/hip_runtime.h>
#include <hip/hip_bf16.h>

typedef float v2f __attribute__((ext_vector_type(2)));
typedef float v8f __attribute__((ext_vector_type(8)));

#define N_NODES 50000
#define N_EDGES 1000000
#define D 64

// ---------------------------------------------------------------------------
// Stage 2: XW = H @ W  with V_WMMA_F32_16X16X4_F32.
// Block = 256 threads = 8 waves; block computes 32 rows x 64 cols.
// Wave (w) tile: Mbase = blockRow + (w>>2)*16, Nbase = (w&3)*16.
// K loop: 64 / 4 = 16 chained WMMAs, accumulator stays in v8f.
// W (64x64 fp32 = 16KB) is staged in LDS once per block.
// ---------------------------------------------------------------------------
__global__ void __launch_bounds__(256)
gcn_gemm_wmma(const float* __restrict__ H, const float* __restrict__ W,
              float* __restrict__ XW, int n) {
    __shared__ float sW[D * D];
    const int tid = threadIdx.x;

    // cooperative 16KB LDS fill (float4, 4 per thread)
    {
        const float4* Wv  = reinterpret_cast<const float4*>(W);
        float4*       sWv = reinterpret_cast<float4*>(sW);
        #pragma unroll
        for (int i = tid; i < (D * D) / 4; i += 256) sWv[i] = Wv[i];
    }
    __syncthreads();

    const int wave = tid >> 5;
    const int lane = tid & 31;
    const int Mbase = blockIdx.x * 32 + (wave >> 2) * 16;
    const int Nbase = (wave & 3) * 16;

    const int mrow  = Mbase + (lane & 15);
    const int mload = (mrow < n) ? mrow : (n - 1);   // clamp so EXEC stays full
    const int koff  = (lane >> 4) * 2;               // A layout: lanes>=16 hold K+2
    const int ncol  = Nbase + (lane & 15);

    const float* __restrict__ hrow = H + (size_t)mload * D;

    v8f c = {};
    #pragma unroll
    for (int k0 = 0; k0 < D; k0 += 4) {
        v2f a;
        a.x = hrow[k0 + koff];
        a.y = hrow[k0 + koff + 1];
        v2f bf;
        bf.x = sW[(k0 + koff) * D + ncol];
        bf.y = sW[(k0 + koff + 1) * D + ncol];
        // D = A(16x4) * B(4x16) + C  -- fp32 WMMA, 8 args
        c = __builtin_amdgcn_wmma_f32_16x16x4_f32(
                /*neg_a=*/false, a, /*neg_b=*/false, bf,
                /*c_mod=*/(short)0, c, /*reuse_a=*/false, /*reuse_b=*/false);
    }

    // C/D layout: VGPR r -> row Mbase + r (+8 for lanes 16..31), col = Nbase + (lane&15)
    const int rsel = (lane >> 4) * 8;
    #pragma unroll
    for (int r = 0; r < 8; ++r) {
        const int row = Mbase + r + rsel;
        if (row < n) XW[(size_t)row * D + ncol] = c[r];
    }
}

// ---------------------------------------------------------------------------
// Stage 3: one wave per edge; lane covers 2 columns (float2) -> coalesced
// 256B gather from L2-resident XW, hardware fp32 atomics into L2-resident AGG.
// ---------------------------------------------------------------------------
__global__ void __launch_bounds__(256)
gcn_edge_scatter(const float* __restrict__ xw, const int* __restrict__ src,
                 const int* __restrict__ dst, const float* __restrict__ w,
                 float* __restrict__ agg, int E) {
    const int gid  = blockIdx.x * blockDim.x + threadIdx.x;
    const int e    = gid >> 5;
    const int lane = gid & 31;
    if (e >= E) return;

    const int   s  = src[e];
    const int   d  = dst[e];
    const float we = w[e];

    const float2 v = reinterpret_cast<const float2*>(xw + (size_t)s * D)[lane];
    float* out = agg + (size_t)d * D + lane * 2;
    unsafeAtomicAdd(out,     we * v.x);
    unsafeAtomicAdd(out + 1, we * v.y);
}

// ---------------------------------------------------------------------------
// Stage 4: h_pre = PReLU(agg + b) (+ residual); accumulate per-feature
// sum / sum-of-squares: registers -> LDS ds_add_f32 -> global fp32 atomics.
// Lane owns feature pair f = 2*(tid&31); 8 nodes per block per sweep.
// ---------------------------------------------------------------------------
__global__ void __launch_bounds__(256)
gcn_post_stats(const float* __restrict__ agg, const float* __restrict__ bias,
               const float* __restrict__ hin, const float* __restrict__ prelu_a,
               int n, int add_residual,
               float* __restrict__ hpre, float* __restrict__ stats) {
    __shared__ float s_sum[D];
    __shared__ float s_sq[D];
    const int tid = threadIdx.x;
    if (tid < D) { s_sum[tid] = 0.f; s_sq[tid] = 0.f; }
    __syncthreads();

    const float a  = prelu_a[0];
    const int   f  = (tid & 31) * 2;
    const float b0 = bias[f], b1 = bias[f + 1];

    float sum0 = 0.f, sum1 = 0.f, sq0 = 0.f, sq1 = 0.f;
    for (int v = blockIdx.x * 8 + (tid >> 5); v < n; v += gridDim.x * 8) {
        const size_t idx = (size_t)v * D + f;
        float h0 = agg[idx]     + b0;
        float h1 = agg[idx + 1] + b1;
        h0 = (h0 >= 0.f) ? h0 : a * h0;
        h1 = (h1 >= 0.f) ? h1 : a * h1;
        if (add_residual) { h0 += hin[idx]; h1 += hin[idx + 1]; }
        hpre[idx]     = h0;
        hpre[idx + 1] = h1;
        sum0 += h0; sum1 += h1;
        sq0  += h0 * h0; sq1 += h1 * h1;
    }
    atomicAdd(&s_sum[f],     sum0);
    atomicAdd(&s_sum[f + 1], sum1);
    atomicAdd(&s_sq[f],      sq0);
    atomicAdd(&s_sq[f + 1],  sq1);
    __syncthreads();

    if (tid < D) {
        unsafeAtomicAdd(&stats[tid],     s_sum[tid]);
        unsafeAtomicAdd(&stats[D + tid], s_sq[tid]);
    }
}

// ---------------------------------------------------------------------------
// Stage 5: batch-norm apply (training-mode batch stats, biased variance).
// ---------------------------------------------------------------------------
__global__ void __launch_bounds__(256)
gcn_bn_apply(const float* __restrict__ hpre, const float* __restrict__ stats,
             const float* __restrict__ gamma, const float* __restrict__ beta,
             float* __restrict__ out, int n) {
    const int gid = blockIdx.x * blockDim.x + threadIdx.x;
    if (gid >= n * D) return;
    const int f = gid & (D - 1);
    const float invn = 1.0f / (float)n;
    const float mean = stats[f] * invn;
    const float var  = stats[D + f] * invn - mean * mean;
    const float inv  = rsqrtf(var + 1e-5f);
    out[gid] = gamma[f] * (hpre[gid] - mean) * inv + beta[f];
}

// ---------------------------------------------------------------------------
extern "C" void kernel_launch(void* const* d_in, const int* in_sizes, int n_in,
                              void* d_out, int out_size, void* d_ws, size_t ws_size,
                              hipStream_t stream) {
    const float* x       = (const float*)d_in[0];
    const int*   esrc    = (const int*)  d_in[1];
    const int*   edst    = (const int*)  d_in[2];
    const float* ew      = (const float*)d_in[3];
    const float* W       = (const float*)d_in[4];
    const float* b       = (const float*)d_in[5];
    const float* prelu_a = (const float*)d_in[6];
    const float* gamma   = (const float*)d_in[7];
    const float* beta    = (const float*)d_in[8];

    const int n = N_NODES, E = N_EDGES;
    const size_t matBytes = (size_t)n * D * sizeof(float);   // 12.8 MB

    char*  ws    = (char*)d_ws;
    float* XW    = (float*)(ws);
    float* AGG   = (float*)(ws + matBytes);
    float* HPRE  = (float*)(ws + 2 * matBytes);
    float* A     = (float*)(ws + 3 * matBytes);
    float* stats = (float*)(ws + 4 * matBytes);              // 128 floats

    const int gemm_blocks = (n + 31) / 32;
    const int edge_blocks = (E * 32 + 255) / 256;
    const int bn_blocks   = (n * D + 255) / 256;

    for (int i = 0; i < 3; ++i) {
        const float* Hin  = (i == 0) ? x : A;
        float*       Hout = (i == 2) ? (float*)d_out : A;

        hipMemsetAsync(AGG,   0, matBytes, stream);
        hipMemsetAsync(stats, 0, 2 * D * sizeof(float), stream);

        gcn_gemm_wmma<<<gemm_blocks, 256, 0, stream>>>(Hin, W + (size_t)i * D * D, XW, n);
        gcn_edge_scatter<<<edge_blocks, 256, 0, stream>>>(XW, esrc, edst, ew, AGG, E);
        gcn_post_stats<<<512, 256, 0, stream>>>(AGG, b + i * D, Hin, prelu_a + i,
                                                n, (i > 0) ? 1 : 0, HPRE, stats);
        gcn_bn_apply<<<bn_blocks, 256, 0, stream>>>(HPRE, stats, gamma + i * D,
                                                    beta + i * D, Hout, n);
    }
}